// WirelessChannel_8976481648737
// MI455X (gfx1250) — compile-verified
//
#include <hip/hip_runtime.h>
#include <math.h>

// ---------------------------------------------------------------------------
// WirelessChannel on MI455X (gfx1250).
// Memory-bound streaming kernel (~201 MB @ 23.3 TB/s => ~8.6 us floor).
// Uses CDNA5 async global->LDS copies (ASYNCcnt) to stage x tiles + conv halo,
// so each x element is read from HBM exactly once and rotated exactly once.
// ---------------------------------------------------------------------------

#define NT    256                  // threads per block (8 wave32)
#define TILE  1024                 // output samples per block (N % TILE == 0)
#define HALO  8                    // halo floats each side (covers conv pad 2, keeps 16B align)
#define REG   (TILE + 2 * HALO)    // 1040 floats staged per component
#define NCH   (REG / 4)            // 260 b128 chunks

typedef float v4f __attribute__((ext_vector_type(4)));

// Generic pointer to __shared__ has the LDS byte offset in its low 32 bits
// (flat LDS aperture: addr[31:0] is the wave-relative LDS address).
__device__ __forceinline__ unsigned lds_off(const void* p) {
    return (unsigned)(size_t)p;
}

// CDNA5 async copy: per-lane 16B global -> LDS, tracked by ASYNCcnt.
// Inline asm (not the clang builtin) so it is portable across ROCm 7.2 /
// amdgpu-toolchain, per CDNA5_HIP.md.
__device__ __forceinline__ void async_load_b128_to_lds(unsigned lds_byte_off,
                                                       const void* gaddr) {
    asm volatile("global_load_async_to_lds_b128 %0, %1, off"
                 :
                 : "v"(lds_byte_off), "v"((unsigned long long)(size_t)gaddr)
                 : "memory");
}

__device__ __forceinline__ void wait_asynccnt0() {
    asm volatile("s_wait_asynccnt 0" ::: "memory");
}

__global__ __launch_bounds__(NT) void wireless_channel_kernel(
    const float* __restrict__ x_real,   const float* __restrict__ x_imag,
    const float* __restrict__ ebno_db,  const float* __restrict__ cfo_u,
    const float* __restrict__ taps_real,const float* __restrict__ taps_imag,
    const float* __restrict__ noise_real,const float* __restrict__ noise_imag,
    float* __restrict__ out, int B, int N)
{
    __shared__ __align__(16) float sxr[REG];
    __shared__ __align__(16) float sxi[REG];

    const int tid        = threadIdx.x;
    const int row        = blockIdx.y;
    const int tileStart  = blockIdx.x * TILE;
    const int regionStart = tileStart - HALO;
    const size_t rowOff  = (size_t)row * (size_t)N;

    // ---- per-row (uniform) parameters: end up in SGPRs -------------------
    const float cfo    = cfo_u[row];
    // phase[n] = 2*pi * ((2u-1)*0.05*15000) * n / 30.72e6  =>  w * n
    const float w      = (cfo * 2.0f - 1.0f) * 750.0f
                         * (6.28318530717958647692f / 30720000.0f);
    const float snr_db = ebno_db[row] + 1.7609125905568124f;  // 10*log10(1.5)
    const float scale  = 0.5f * exp10f(-snr_db * 0.05f);      // noise_std / 2
    float tr[5], ti[5];
#pragma unroll
    for (int k = 0; k < 5; ++k) {
        tr[k] = taps_real[row * 5 + k];
        ti[k] = taps_imag[row * 5 + k];
    }

    // ---- stage 0: async-copy raw x tile (+halo) into LDS -----------------
    // chunk c covers logical sample indices regionStart + 4c .. +3.
    // Clamp global address into range (OOB chunks are fully zeroed later).
    {
        const int c   = tid;
        const int idx = regionStart + c * 4;
        const int cl  = idx < 0 ? 0 : (idx > N - 4 ? N - 4 : idx);
        async_load_b128_to_lds(lds_off(&sxr[c * 4]), x_real + rowOff + cl);
        async_load_b128_to_lds(lds_off(&sxi[c * 4]), x_imag + rowOff + cl);
    }
    if (tid < NCH - NT) {  // 4 extra chunks to cover both halos
        const int c   = NT + tid;
        const int idx = regionStart + c * 4;
        const int cl  = idx < 0 ? 0 : (idx > N - 4 ? N - 4 : idx);
        async_load_b128_to_lds(lds_off(&sxr[c * 4]), x_real + rowOff + cl);
        async_load_b128_to_lds(lds_off(&sxi[c * 4]), x_imag + rowOff + cl);
    }
    wait_asynccnt0();  // this wave's LDS chunks are now resident

    // ---- stage 1: CFO rotation in place (each thread owns its chunks) ----
    auto rotate_chunk = [&](int c) {
        const int idx0 = regionStart + c * 4;
#pragma unroll
        for (int j = 0; j < 4; ++j) {
            const int g = idx0 + j;                // logical sample index
            float xr = 0.0f, xi = 0.0f;
            if ((unsigned)g < (unsigned)N) {       // zero-pad outside [0,N)
                xr = sxr[c * 4 + j];
                xi = sxi[c * 4 + j];
            }
            float s, cz;
            __sincosf(w * (float)g, &s, &cz);
            sxr[c * 4 + j] = xr * cz - xi * s;
            sxi[c * 4 + j] = xr * s + xi * cz;
        }
    };
    rotate_chunk(tid);
    if (tid < NCH - NT) rotate_chunk(NT + tid);
    __syncthreads();  // rotated tile visible to all waves

    // ---- stage 2: 5-tap complex depthwise conv + AWGN, vectorized I/O ----
    const int n0 = tileStart + tid * 4;
    if (n0 < N) {
        const int l0 = tid * 4 + HALO;  // local index of n0 inside region
        float vr[8], vi[8];
#pragma unroll
        for (int m = 0; m < 8; ++m) {   // window for 4 outputs: [n0-2, n0+5]
            vr[m] = sxr[l0 - 2 + m];
            vi[m] = sxi[l0 - 2 + m];
        }

        if (n0 + 4 <= N) {
            const v4f nr = __builtin_nontemporal_load(
                (const v4f*)(noise_real + rowOff + n0));
            const v4f ni = __builtin_nontemporal_load(
                (const v4f*)(noise_imag + rowOff + n0));
            v4f oyr, oyi;
#pragma unroll
            for (int j = 0; j < 4; ++j) {
                float crr = 0.f, cir = 0.f, cri = 0.f, cii = 0.f;
#pragma unroll
                for (int k = 0; k < 5; ++k) {
                    crr = fmaf(tr[k], vr[j + k], crr);
                    cir = fmaf(tr[k], vi[j + k], cir);
                    cri = fmaf(ti[k], vr[j + k], cri);
                    cii = fmaf(ti[k], vi[j + k], cii);
                }
                oyr[j] = (crr - cii) + nr[j] * scale;
                oyi[j] = (cri + cir) + ni[j] * scale;
            }
            __builtin_nontemporal_store(oyr, (v4f*)(out + rowOff + n0));
            __builtin_nontemporal_store(
                oyi, (v4f*)(out + (size_t)B * (size_t)N + rowOff + n0));
        } else {
            // generic tail (not taken for N=131072, TILE=1024)
            for (int j = 0; j < 4 && n0 + j < N; ++j) {
                float crr = 0.f, cir = 0.f, cri = 0.f, cii = 0.f;
                for (int k = 0; k < 5; ++k) {
                    crr = fmaf(tr[k], vr[j + k], crr);
                    cir = fmaf(tr[k], vi[j + k], cir);
                    cri = fmaf(ti[k], vr[j + k], cri);
                    cii = fmaf(ti[k], vi[j + k], cii);
                }
                const float nrj = noise_real[rowOff + n0 + j];
                const float nij = noise_imag[rowOff + n0 + j];
                out[rowOff + n0 + j] = (crr - cii) + nrj * scale;
                out[(size_t)B * (size_t)N + rowOff + n0 + j] =
                    (cri + cir) + nij * scale;
            }
        }
    }
}

extern "C" void kernel_launch(void* const* d_in, const int* in_sizes, int n_in,
                              void* d_out, int out_size, void* d_ws, size_t ws_size,
                              hipStream_t stream) {
    (void)n_in; (void)out_size; (void)d_ws; (void)ws_size;
    const float* x_real     = (const float*)d_in[0];
    const float* x_imag     = (const float*)d_in[1];
    const float* ebno_db    = (const float*)d_in[2];
    const float* cfo_u      = (const float*)d_in[3];
    const float* taps_real  = (const float*)d_in[4];
    const float* taps_imag  = (const float*)d_in[5];
    const float* noise_real = (const float*)d_in[6];
    const float* noise_imag = (const float*)d_in[7];
    float* out = (float*)d_out;

    const int B = in_sizes[2];            // ebno_db has B elements
    const int N = in_sizes[0] / B;        // x_real is B*N

    dim3 grid((N + TILE - 1) / TILE, B);  // 128 x 64 for the given shapes
    hipLaunchKernelGGL(wireless_channel_kernel, grid, dim3(NT), 0, stream,
                       x_real, x_imag, ebno_db, cfo_u, taps_real, taps_imag,
                       noise_real, noise_imag, out, B, N);
}